// MFA_13778255086333
// MI455X (gfx1250) — compile-verified
//
#include <hip/hip_runtime.h>
#include <math.h>

typedef __attribute__((ext_vector_type(2))) float v2f;
typedef __attribute__((ext_vector_type(8))) float v8f;

#define Dd 256
#define Kk 32
#define Qq 16
#define Nn 8192

// workspace layout (in floats)
#define OFF_AT    0                       // [544][256]: rows 0..511 = A_k^T (q rows), rows 512..543 = w_k rows
#define ROWS_AT   (Kk*Qq + Kk)            // 544
#define OFF_MINV  (ROWS_AT*Dd)            // 139264 : [K][16][16]
#define OFF_B     (OFF_MINV + Kk*Qq*Qq)   // 147456 : b_k[q] = A_k^T mu_k
#define OFF_CONST (OFF_B + Kk*Qq)         // 147968 : log_pi_k - 0.5*(D log2pi + logdetM + sum log psi)
#define OFF_C     (OFF_CONST + Kk)        // 148000 : mu^T Psi^-1 mu
#define OFF_PSI   (OFF_C + Kk)            // 148032 : psi^-1 (256)
#define WS_FLOATS (OFF_PSI + Dd)          // 148288 floats ~ 593 KB

__device__ __forceinline__ v8f wmma_f32_16x16x4(v2f a, v2f b, v8f c) {
  // 8 args: (neg_a, A, neg_b, B, c_mod, C, reuse_a, reuse_b)
  return __builtin_amdgcn_wmma_f32_16x16x4_f32(false, a, false, b, (short)0, c,
                                               false, false);
}

// ---------------- Stage 1: per-component precompute (K blocks x 256 thr) ----
__global__ __launch_bounds__(256) void mfa_setup_kernel(
    const float* __restrict__ log_pi, const float* __restrict__ mu,
    const float* __restrict__ Lambda, const float* __restrict__ log_psi,
    float* __restrict__ ws) {
  const int k = blockIdx.x;
  const int t = threadIdx.x;
  __shared__ float pinv[Dd];
  __shared__ float red[Dd];
  __shared__ float Msh[Qq][Qq];
  __shared__ float sumlog_sh, c_sh;

  // effective diagonal: exp(log_psi) + 1e-6 + 1e-5
  float pe = __expf(log_psi[t]) + 1.1e-5f;
  float pv = 1.0f / pe;
  pinv[t] = pv;
  red[t]  = __logf(pe);
  __syncthreads();
  for (int s = 128; s > 0; s >>= 1) {
    if (t < s) red[t] += red[t + s];
    __syncthreads();
  }
  if (t == 0) sumlog_sh = red[0];
  __syncthreads();

  float muv = mu[k * Dd + t];
  float wv  = pv * muv;
#pragma unroll
  for (int q = 0; q < Qq; ++q)
    ws[OFF_AT + (k * Qq + q) * Dd + t] = Lambda[(k * Dd + t) * Qq + q] * pv;
  ws[OFF_AT + (Kk * Qq + k) * Dd + t] = wv;   // w row
  if (k == 0) ws[OFF_PSI + t] = pv;

  // c = mu^T Psi^-1 mu
  red[t] = wv * muv;
  __syncthreads();
  for (int s = 128; s > 0; s >>= 1) {
    if (t < s) red[t] += red[t + s];
    __syncthreads();
  }
  if (t == 0) c_sh = red[0];

  // M = I + Lambda^T Psi^-1 Lambda  (one entry per thread)
  {
    int q = t >> 4, p = t & 15;
    float s = 0.f;
    for (int d = 0; d < Dd; ++d)
      s += Lambda[(k * Dd + d) * Qq + q] * pinv[d] * Lambda[(k * Dd + d) * Qq + p];
    Msh[q][p] = s + (q == p ? 1.0f : 0.0f);
  }
  // b_q = sum_d A[d][q] mu[d]
  if (t < Qq) {
    float s = 0.f;
    for (int d = 0; d < Dd; ++d)
      s += Lambda[(k * Dd + d) * Qq + t] * pinv[d] * mu[k * Dd + d];
    ws[OFF_B + k * Qq + t] = s;
  }
  __syncthreads();

  if (t == 0) {
    // 16x16 Cholesky of M in shared
    float logdet = 0.f;
    for (int j = 0; j < Qq; ++j) {
      float s = Msh[j][j];
      for (int i = 0; i < j; ++i) s -= Msh[j][i] * Msh[j][i];
      float ljj = sqrtf(s);
      Msh[j][j] = ljj;
      logdet += 2.0f * __logf(ljj);
      for (int r = j + 1; r < Qq; ++r) {
        float s2 = Msh[r][j];
        for (int i = 0; i < j; ++i) s2 -= Msh[r][i] * Msh[j][i];
        Msh[r][j] = s2 / ljj;
      }
    }
    // M^-1 via triangular solves
    for (int e = 0; e < Qq; ++e) {
      float y[Qq], xv[Qq];
      for (int j = 0; j < Qq; ++j) {
        float s = (j == e) ? 1.0f : 0.0f;
        for (int i = 0; i < j; ++i) s -= Msh[j][i] * y[i];
        y[j] = s / Msh[j][j];
      }
      for (int j = Qq - 1; j >= 0; --j) {
        float s = y[j];
        for (int i = j + 1; i < Qq; ++i) s -= Msh[i][j] * xv[i];
        xv[j] = s / Msh[j][j];
      }
      for (int j = 0; j < Qq; ++j)
        ws[OFF_MINV + k * Qq * Qq + j * Qq + e] = xv[j];
    }
    const float LOG2PI = 1.8378770664093453f;
    ws[OFF_CONST + k] =
        log_pi[k] - 0.5f * ((float)Dd * LOG2PI + logdet + sumlog_sh);
    ws[OFF_C + k] = c_sh;
  }
}

// ---------------- Stage 2: fused WMMA GEMM + Woodbury epilogue --------------
// grid 64 blocks x 256 threads; each wave32 owns 16 samples.
__global__ __launch_bounds__(256) void mfa_main_kernel(
    const float* __restrict__ X, const float* __restrict__ ws,
    float* __restrict__ out) {
  const int l   = threadIdx.x & 31;
  const int wid = threadIdx.x >> 5;
  const int hi  = l >> 4;
  const int lo  = l & 15;
  const int n0  = (blockIdx.x * 8 + wid) * 16;

  __shared__ float zs[8][Qq][17];    // per-wave Z tile (q x n)
  __shared__ float xws[8][Kk][17];   // per-wave x^T w_k (k x n)

  const float* AT = ws + OFF_AT;
  const float* pX = X + (size_t)(n0 + lo) * Dd + 2 * hi;
  __builtin_prefetch(pX, 0, 0);

  // Phase A: xw tiles (A rows 512..543, row index = k)
  v8f acc0 = {}; v8f acc1 = {};
  {
    const float* pA0 = AT + (size_t)(Kk * Qq + lo) * Dd + 2 * hi;
    const float* pA1 = AT + (size_t)(Kk * Qq + 16 + lo) * Dd + 2 * hi;
#pragma unroll 8
    for (int d0 = 0; d0 < Dd; d0 += 4) {
      v2f b  = *(const v2f*)(pX + d0);
      v2f a0 = *(const v2f*)(pA0 + d0);
      v2f a1 = *(const v2f*)(pA1 + d0);
      acc0 = wmma_f32_16x16x4(a0, b, acc0);
      acc1 = wmma_f32_16x16x4(a1, b, acc1);
    }
  }
#pragma unroll
  for (int v = 0; v < 8; ++v) {
    xws[wid][v + 8 * hi][lo]      = acc0[v];
    xws[wid][16 + v + 8 * hi][lo] = acc1[v];
  }

  // s_x = x^T Psi^-1 x for this lane's sample
  float sx = 0.f;
  {
    const float4* xr = (const float4*)(X + (size_t)(n0 + lo) * Dd);
    const float4* pv = (const float4*)(ws + OFF_PSI);
#pragma unroll 4
    for (int i = 0; i < Dd / 4; ++i) {
      float4 x = xr[i], p = pv[i];
      sx += x.x * x.x * p.x + x.y * x.y * p.y + x.z * x.z * p.z + x.w * x.w * p.w;
    }
  }

  float mrun = -INFINITY, srun = 0.f;

  for (int k = 0; k < Kk; ++k) {
    // z-tile: Z = A_k^T X  (16 q-rows x 16 samples), K-loop over d in steps of 4
    v8f accz = {};
    const float* pAz = AT + (size_t)(k * Qq + lo) * Dd + 2 * hi;
#pragma unroll 8
    for (int d0 = 0; d0 < Dd; d0 += 4) {
      v2f b = *(const v2f*)(pX + d0);
      v2f a = *(const v2f*)(pAz + d0);
      accz = wmma_f32_16x16x4(a, b, accz);
    }
    // z -= b_k  (D-layout: vgpr v, half hi -> q = v + 8*hi)
#pragma unroll
    for (int v = 0; v < 8; ++v) accz[v] -= ws[OFF_B + k * Qq + v + 8 * hi];
    // spill Z to LDS so it can be re-read in B-operand layout
#pragma unroll
    for (int v = 0; v < 8; ++v) zs[wid][v + 8 * hi][lo] = accz[v];
    asm volatile("s_wait_dscnt 0" ::: "memory");

    // U = Minv_k * Z via 4 WMMAs (K=16 in steps of 4)
    v8f accu = {};
    const float* Mi = ws + OFF_MINV + k * Qq * Qq;
#pragma unroll
    for (int kk = 0; kk < Qq; kk += 4) {
      v2f aM = *(const v2f*)(Mi + lo * Qq + kk + 2 * hi);
      v2f bZ;
      bZ[0] = zs[wid][kk + 2 * hi + 0][lo];
      bZ[1] = zs[wid][kk + 2 * hi + 1][lo];
      accu = wmma_f32_16x16x4(aM, bZ, accu);
    }
    // corr(n) = sum_q Z[q][n] * U[q][n]; halves combined across lanes n / n+16
    float part = 0.f;
#pragma unroll
    for (int v = 0; v < 8; ++v) part += accz[v] * accu[v];
    float corr = part + __shfl_xor(part, 16, 32);

    float xw = xws[wid][k][lo];
    float q0 = sx - 2.0f * xw + ws[OFF_C + k];          // diff^T Psi^-1 diff
    float lr = ws[OFF_CONST + k] - 0.5f * (q0 - corr);  // log_pi + log_prob
    if (l < 16) out[(size_t)(n0 + lo) * Kk + k] = lr;

    // online logsumexp
    float nm = fmaxf(mrun, lr);
    srun = srun * __expf(mrun - nm) + __expf(lr - nm);
    mrun = nm;
  }
  if (l < 16) out[(size_t)Nn * Kk + n0 + lo] = mrun + __logf(srun);
}

// ---------------- Stage 3: normalize log responsibilities -------------------
__global__ void mfa_norm_kernel(float* __restrict__ out) {
  int i = blockIdx.x * blockDim.x + threadIdx.x;
  if (i < Nn * Kk) {
    int n = i >> 5;
    out[i] = out[i] - out[Nn * Kk + n];
  }
}

extern "C" void kernel_launch(void* const* d_in, const int* in_sizes, int n_in,
                              void* d_out, int out_size, void* d_ws,
                              size_t ws_size, hipStream_t stream) {
  const float* X       = (const float*)d_in[0];
  const float* log_pi  = (const float*)d_in[1];
  const float* mu      = (const float*)d_in[2];
  const float* Lambda  = (const float*)d_in[3];
  const float* log_psi = (const float*)d_in[4];
  float* ws  = (float*)d_ws;
  float* out = (float*)d_out;
  (void)in_sizes; (void)n_in; (void)out_size; (void)ws_size;

  mfa_setup_kernel<<<Kk, Dd, 0, stream>>>(log_pi, mu, Lambda, log_psi, ws);
  mfa_main_kernel<<<Nn / 128, 256, 0, stream>>>(X, ws, out);
  mfa_norm_kernel<<<(Nn * Kk + 255) / 256, 256, 0, stream>>>(out);
}